// GHMcLoss_38671885533680
// MI455X (gfx1250) — compile-verified
//
#include <hip/hip_runtime.h>

#define NBINS 10
#define BLOCK_THREADS 256
#define MAIN_BLOCKS 2048
#define WS_STRIDE 32   // floats per block in workspace: [0..9]=S_b, [10..19]=count_b, rest 0
#define HSTRIDE 21     // 20 live slots + 1 pad word; 21 coprime with 64 banks

typedef __attribute__((ext_vector_type(2))) float v2f;
typedef __attribute__((ext_vector_type(8))) float v8f;

// Pass 1: one pass over pred; per-thread-private LDS bins updated with native
// ds_add_f32 (1 DS issue per update, zero same-address conflicts by construction).
__global__ __launch_bounds__(BLOCK_THREADS) void ghmc_partials(
    const float* __restrict__ pred, const int* __restrict__ target,
    float* __restrict__ ws, int nquads, int cshift, int cmask) {
  __shared__ float hist[BLOCK_THREADS * HSTRIDE];   // ~21.5 KB of 320 KB/WGP

  float* mybins = &hist[threadIdx.x * HSTRIDE];
#pragma unroll
  for (int k = 0; k < 2 * NBINS; ++k) mybins[k] = 0.0f;
  __syncthreads();

  const float4* pred4 = (const float4*)pred;
  const int stride = blockDim.x * gridDim.x;
  for (int v = blockIdx.x * blockDim.x + threadIdx.x; v < nquads; v += stride) {
    float4 p4 = pred4[v];
    const int i0   = v << 2;
    const int row  = i0 >> cshift;   // C is a power of two (512)
    const int col0 = i0 & cmask;     // 4 consecutive cols never cross a row (4 | C)
    const int tgt  = target[row];

    float pv[4] = {p4.x, p4.y, p4.z, p4.w};
#pragma unroll
    for (int j = 0; j < 4; ++j) {
      float p = pv[j];
      float t = (tgt == (col0 + j)) ? 1.0f : 0.0f;
      float e = __expf(-p);                         // v_exp_f32
      float s = __builtin_amdgcn_rcpf(1.0f + e);    // sigmoid(p)
      float g = fabsf(s - t);
      int   b = (int)(g * 10.0f);
      b = b > (NBINS - 1) ? (NBINS - 1) : b;
      // log1p(exp(-|p|)) == -log(sigmoid(|p|)); sigmoid(|p|) = p>=0 ? s : 1-s
      float sAbs = (p >= 0.0f) ? s : (1.0f - s);
      float bce  = fmaxf(p, 0.0f) - p * t - __logf(sAbs);

      // Two ds_add_f32 into this thread's private region; second uses the
      // same base with a constant element offset (ds immediate offset).
      (void)__hip_atomic_fetch_add(&mybins[b], bce, __ATOMIC_RELAXED,
                                   __HIP_MEMORY_SCOPE_WORKGROUP);
      (void)__hip_atomic_fetch_add(&mybins[NBINS + b], 1.0f, __ATOMIC_RELAXED,
                                   __HIP_MEMORY_SCOPE_WORKGROUP);
    }
  }
  __syncthreads();

  // Column-reduce the 256 private regions (one wave; 20 consecutive addresses
  // per row -> conflict-free), one 32-float store per block, no global atomics.
  if (threadIdx.x < WS_STRIDE) {
    float acc = 0.0f;
    if (threadIdx.x < 2 * NBINS) {
      for (int r = 0; r < BLOCK_THREADS; ++r)
        acc += hist[r * HSTRIDE + threadIdx.x];
    }
    ws[(size_t)blockIdx.x * WS_STRIDE + threadIdx.x] = acc;  // cols 20..31 = 0
  }
}

// Pass 2: column-sum the nblocks x 32 partial matrix with WMMA, then finalize.
// D = A(16x4, all ones) x B(4x16 slab) + C  ==> every row of D is the column sum.
__global__ void ghmc_finalize(const float* __restrict__ ws,
                              float* __restrict__ out, int nblocks) {
  const int lane = threadIdx.x;      // launched with exactly 32 threads (EXEC all-1s)
  const int half = lane >> 4;
  const int ln   = lane & 15;
  __shared__ float colsum[WS_STRIDE];

#if __has_builtin(__builtin_amdgcn_wmma_f32_16x16x4_f32)
  v2f ones; ones.x = 1.0f; ones.y = 1.0f;   // A(16x4): every element 1.0
  v8f acc0 = {};                            // column tile 0..15
  v8f acc1 = {};                            // column tile 16..31
  int rbase = 0;
  for (; rbase + 4 <= nblocks; rbase += 4) {
    // B layout (mirrors documented A/C striping): VGPR j -> K=j (lanes 0-15),
    // K=j+2 (lanes 16-31); so this lane's rows are r0, r0+1.
    const int r0 = rbase + 2 * half;
    v2f b0, b1;
    b0.x = ws[(size_t)(r0 + 0) * WS_STRIDE + ln];
    b0.y = ws[(size_t)(r0 + 1) * WS_STRIDE + ln];
    b1.x = ws[(size_t)(r0 + 0) * WS_STRIDE + 16 + ln];
    b1.y = ws[(size_t)(r0 + 1) * WS_STRIDE + 16 + ln];
    acc0 = __builtin_amdgcn_wmma_f32_16x16x4_f32(false, ones, false, b0,
                                                 (short)0, acc0, false, false);
    acc1 = __builtin_amdgcn_wmma_f32_16x16x4_f32(false, ones, false, b1,
                                                 (short)0, acc1, false, false);
  }
  // Scalar tail for nblocks % 4 rows.
  float extra0 = 0.f, extra1 = 0.f;
  for (int r = rbase; r < nblocks; ++r) {
    extra0 += ws[(size_t)r * WS_STRIDE + ln];
    extra1 += ws[(size_t)r * WS_STRIDE + 16 + ln];
  }
  if (lane < 16) {                 // D VGPR0 lanes 0-15 = row 0, cols 0-15
    colsum[ln]      = acc0[0] + extra0;
    colsum[16 + ln] = acc1[0] + extra1;
  }
#else
  float s = 0.f;
  for (int r = 0; r < nblocks; ++r) s += ws[(size_t)r * WS_STRIDE + lane];
  colsum[lane] = s;
#endif
  __syncthreads();

  if (threadIdx.x == 0) {
    float acc = 0.f;
    int n = 0;
#pragma unroll
    for (int b = 0; b < NBINS; ++b) {
      float cnt = colsum[NBINS + b];
      float sb  = colsum[b];
      if (cnt > 0.f) { acc += sb / cnt; ++n; }
    }
    out[0] = (acc / (float)(n > 0 ? n : 1)) * 1.0f;  // LOSS_WEIGHT = 1
  }
}

extern "C" void kernel_launch(void* const* d_in, const int* in_sizes, int n_in,
                              void* d_out, int out_size, void* d_ws, size_t ws_size,
                              hipStream_t stream) {
  const float* pred   = (const float*)d_in[0];
  const int*   target = (const int*)d_in[1];
  float*       ws     = (float*)d_ws;
  float*       out    = (float*)d_out;

  const int NC = in_sizes[0];          // 131072 * 512
  const int N  = in_sizes[1];          // 131072
  const int C  = NC / N;               // 512 (power of two)
  int cshift = 0; while ((1 << cshift) < C) ++cshift;
  const int cmask  = C - 1;
  const int nquads = NC / 4;

  int nb = MAIN_BLOCKS;
  const size_t need = (size_t)nb * WS_STRIDE * sizeof(float);
  if (need > ws_size) {
    nb = (int)(ws_size / (WS_STRIDE * sizeof(float)));
    if (nb < 1) nb = 1;
  }

  ghmc_partials<<<nb, BLOCK_THREADS, 0, stream>>>(pred, target, ws, nquads,
                                                  cshift, cmask);
  ghmc_finalize<<<1, 32, 0, stream>>>(ws, out, nb);
}